// PairwiseMLPSimilarity_3848290697817
// MI455X (gfx1250) — compile-verified
//
#include <hip/hip_runtime.h>
#include <math.h>

#define B_    8
#define N_    512
#define E_    64
#define H_    128
#define MTOT  (B_ * N_)          // 4096 rows
#define NTILE (N_ / 32)          // 16 tiles per dim
#define NTRI  (NTILE * (NTILE + 1) / 2)  // 136 upper tile-pairs

typedef float v2f __attribute__((ext_vector_type(2)));
typedef float v8f __attribute__((ext_vector_type(8)));

// ---------------------------------------------------------------------------
// Kernel 1: A = X*Wc1[:64,:], Bm = X*Wc1[64:,:], D = X*Ws1  via f32 WMMA.
// One wave per 16x16 tile. 4096 rows x 384 cols -> 256 * 24 = 6144 tiles.
// ---------------------------------------------------------------------------
__global__ __launch_bounds__(256) void precompute_wmma(
    const float* __restrict__ X,    // (4096, 64)
    const float* __restrict__ Wc1,  // (128, 128)
    const float* __restrict__ Ws1,  // (64, 128)
    float* __restrict__ ws)         // A | Bm | D, each 4096*128 floats
{
  const int wave = threadIdx.x >> 5;
  const int lane = threadIdx.x & 31;
  const int tile = blockIdx.x * 8 + wave;      // 0..6143, exact coverage
  const int tile_m = tile / 24;                // row tile (16 rows)
  const int tile_c = tile % 24;                // col tile over 3*128 cols
  const int w      = tile_c >> 3;              // 0: Wc1_top, 1: Wc1_bot, 2: Ws1
  const int hcol0  = (tile_c & 7) << 4;

  const float* Wp = (w == 2) ? Ws1 : Wc1;
  const int krowbase = (w == 1) ? 64 : 0;

  // A-fragment: lanes 0-15 -> M=0..15 with K={k0,k0+1}; lanes 16-31 -> K={k0+2,k0+3}
  const int mrow  = tile_m * 16 + (lane & 15);
  const int khalf = (lane >> 4) << 1;          // 0 or 2
  const int col   = hcol0 + (lane & 15);       // B/N column for this lane

  const float* xrow = X + mrow * E_;

  v8f acc = {0.f, 0.f, 0.f, 0.f, 0.f, 0.f, 0.f, 0.f};
#pragma unroll
  for (int k0 = 0; k0 < E_; k0 += 4) {
    v2f a, bfrag;
    a.x = xrow[k0 + khalf];
    a.y = xrow[k0 + khalf + 1];
    const int kr = krowbase + k0 + khalf;
    bfrag.x = Wp[kr * H_ + col];
    bfrag.y = Wp[(kr + 1) * H_ + col];
    // (neg_a, A, neg_b, B, c_mod, C, reuse_a, reuse_b)
    acc = __builtin_amdgcn_wmma_f32_16x16x4_f32(
        false, a, false, bfrag, (short)0, acc, false, false);
  }

  // C/D layout: VGPR r -> M=r (lanes 0-15), M=r+8 (lanes 16-31); N = lane&15
  float* dst = ws + (size_t)w * (MTOT * H_);
  const int mbase = tile_m * 16 + ((lane >> 4) << 3);
#pragma unroll
  for (int r = 0; r < 8; ++r)
    dst[(size_t)(mbase + r) * H_ + col] = acc[r];
}

// ---------------------------------------------------------------------------
// Kernel 2: pair stage. sim(i,j) = sigmoid(sum_h relu(A_i+B_j+bc1)*Wc2 + bc2)
// One workgroup per (batch, upper-triangular 32x32 tile). 2x2 register block.
// ---------------------------------------------------------------------------
#define LPAD 132  // 132 % 64 == 4 -> conflict-free strided column reads

__global__ __launch_bounds__(256) void pair_kernel(
    const float* __restrict__ ws,   // A | Bm
    const float* __restrict__ bc1,  // (128)
    const float* __restrict__ Wc2,  // (128)
    const float* __restrict__ bc2,  // (1)
    float* __restrict__ out)        // (8, 512, 512)
{
  __shared__ float sA[32 * LPAD];
  __shared__ float sB[32 * LPAD];
  __shared__ float w2s[H_];

  const int tid = threadIdx.x;
  const int b   = blockIdx.x / NTRI;
  int t         = blockIdx.x % NTRI;
  int ti = 0;
  while (t >= NTILE - ti) { t -= NTILE - ti; ++ti; }   // uniform scalar decode
  const int tj = ti + t;                               // ti <= tj
  const int i0 = ti * 32, j0 = tj * 32;

  const float* Arow = ws + ((size_t)b * N_ + i0) * H_;
  const float* Brow = ws + (size_t)MTOT * H_ + ((size_t)b * N_ + j0) * H_;

  for (int s = tid; s < 32 * H_; s += 256) {
    const int i = s >> 7, h = s & 127;
    sA[i * LPAD + h] = Arow[(size_t)i * H_ + h] + bc1[h];  // fold bias into A
    sB[i * LPAD + h] = Brow[(size_t)i * H_ + h];
  }
  if (tid < H_) w2s[tid] = Wc2[tid];
  __syncthreads();

  const int tx = tid & 15;   // j sub-block
  const int ty = tid >> 4;   // i sub-block
  const float* pa0 = sA + (ty * 2) * LPAD;
  const float* pa1 = pa0 + LPAD;
  const float* pb0 = sB + (tx * 2) * LPAD;
  const float* pb1 = pb0 + LPAD;

  float a00 = 0.f, a01 = 0.f, a10 = 0.f, a11 = 0.f;
#pragma unroll 8
  for (int h = 0; h < H_; ++h) {
    const float wv = w2s[h];
    const float x0 = pa0[h], x1 = pa1[h];
    const float y0 = pb0[h], y1 = pb1[h];
    a00 = fmaf(fmaxf(x0 + y0, 0.f), wv, a00);
    a01 = fmaf(fmaxf(x0 + y1, 0.f), wv, a01);
    a10 = fmaf(fmaxf(x1 + y0, 0.f), wv, a10);
    a11 = fmaf(fmaxf(x1 + y1, 0.f), wv, a11);
  }

  const float c2 = bc2[0];
  float* ob = out + (size_t)b * N_ * N_;
  const int ig = i0 + ty * 2, jg = j0 + tx * 2;
  const float v[2][2] = {{a00, a01}, {a10, a11}};
#pragma unroll
  for (int r = 0; r < 2; ++r)
#pragma unroll
    for (int c = 0; c < 2; ++c) {
      const int i = ig + r, j = jg + c;
      if (j > i) {  // strict upper triangle; diag handled separately
        const float s = 1.0f / (1.0f + __expf(-(v[r][c] + c2)));
        ob[(size_t)i * N_ + j] = s;
        ob[(size_t)j * N_ + i] = s;
      }
    }
}

// ---------------------------------------------------------------------------
// Kernel 3: diagonal. sim(n,n) = sigmoid(sum_h relu(D+bs1)*Ws2 + bs2)
// ---------------------------------------------------------------------------
__global__ __launch_bounds__(256) void diag_kernel(
    const float* __restrict__ D,    // (4096, 128)
    const float* __restrict__ bs1,  // (128)
    const float* __restrict__ Ws2,  // (128)
    const float* __restrict__ bs2,  // (1)
    float* __restrict__ out)
{
  const int m = blockIdx.x * 256 + threadIdx.x;  // 0..4095
  const int b = m >> 9, n = m & 511;
  const float* drow = D + (size_t)m * H_;
  float acc = 0.f;
#pragma unroll 4
  for (int h = 0; h < H_; ++h)
    acc = fmaf(fmaxf(drow[h] + bs1[h], 0.f), Ws2[h], acc);
  const float s = 1.0f / (1.0f + __expf(-(acc + bs2[0])));
  out[(size_t)b * N_ * N_ + (size_t)n * N_ + n] = s;
}

// ---------------------------------------------------------------------------
extern "C" void kernel_launch(void* const* d_in, const int* in_sizes, int n_in,
                              void* d_out, int out_size, void* d_ws, size_t ws_size,
                              hipStream_t stream) {
  const float* X   = (const float*)d_in[0];
  const float* Wc1 = (const float*)d_in[1];
  const float* bc1 = (const float*)d_in[2];
  const float* Wc2 = (const float*)d_in[3];
  const float* bc2 = (const float*)d_in[4];
  const float* Ws1 = (const float*)d_in[5];
  const float* bs1 = (const float*)d_in[6];
  const float* Ws2 = (const float*)d_in[7];
  const float* bs2 = (const float*)d_in[8];
  float* out = (float*)d_out;
  float* ws  = (float*)d_ws;  // needs 3 * 4096 * 128 * 4 = 6 MB

  // 6144 tiles / 8 waves per block = 768 blocks
  precompute_wmma<<<768, 256, 0, stream>>>(X, Wc1, Ws1, ws);
  pair_kernel<<<B_ * NTRI, 256, 0, stream>>>(ws, bc1, Wc2, bc2, out);
  diag_kernel<<<MTOT / 256, 256, 0, stream>>>(ws + 2 * (size_t)MTOT * H_,
                                              bs1, Ws2, bs2, out);
}